// GraphAttention_86715389706288
// MI455X (gfx1250) — compile-verified
//
#include <hip/hip_runtime.h>
#include <math.h>

typedef float v2f __attribute__((ext_vector_type(2)));
typedef float v4f __attribute__((ext_vector_type(4)));
typedef float v8f __attribute__((ext_vector_type(8)));

#define ALPHA 0.2f
#define MASK_VAL -9000000000000000.0f

constexpr int Bc = 8, Nc = 1024, Fc = 256, Hc = 4, HSc = 64;   // problem shape
constexpr int OUTc = Hc * HSc;                                  // 256

// ---------------------------------------------------------------------------
// Kernel 1: WhT[bh][d][n] = sum_f x[b,n,f] * W[h,f,d]   (fp32 WMMA 16x16x4)
// One wave per 16x16 tile; stores the D tile TRANSPOSED so that:
//   - stores coalesce into 2x global_store_b128 per lane (r -> consecutive n)
//   - kernel 3's B operand (K-direction = n) becomes contiguous b64 loads
// ---------------------------------------------------------------------------
__global__ __launch_bounds__(256) void k_wh(const float* __restrict__ x,
                                            const float* __restrict__ W,
                                            float* __restrict__ WhT) {
  const int lane = threadIdx.x & 31;
  const int wave = threadIdx.x >> 5;
  const int tile = blockIdx.x * 8 + wave;      // 8192 tiles total
  const int bh   = tile >> 8;                  // / (64 ntiles * 4 dtiles)
  const int rem  = tile & 255;
  const int ntile = rem >> 2;
  const int dtile = rem & 3;
  const int b = bh >> 2, h = bh & 3;
  const int lm   = lane & 15;
  const int koff = (lane < 16) ? 0 : 2;
  const int n0 = ntile * 16;
  const int d0 = dtile * 16;

  // A: x rows (M = n0+lm), consecutive K pair -> one b64 load
  const float* pa = x + (size_t)(b * Nc + n0 + lm) * Fc + koff;
  // B: W rows K (stride HS), column d0+lm
  const float* pb = W + (size_t)(h * Fc + koff) * HSc + d0 + lm;

  v8f c = {};
#pragma unroll 4
  for (int k = 0; k < Fc; k += 4) {
    v2f a = *(const v2f*)(pa + k);
    v2f bm;
    bm.x = pb[k * HSc];
    bm.y = pb[k * HSc + HSc];
    c = __builtin_amdgcn_wmma_f32_16x16x4_f32(false, a, false, bm,
                                              (short)0, c, false, false);
  }

  // D layout: VGPR r -> row n0+rbase+r, lane%16 -> col d0+lm.
  // Transposed store: WhT[bh][d0+lm][n0+rbase + 0..7] = c[0..7] (8 consecutive)
  const int rbase = (lane < 16) ? 0 : 8;
  float* pout = WhT + (size_t)bh * HSc * Nc + (size_t)(d0 + lm) * Nc + n0 + rbase;
  v4f lo = {c[0], c[1], c[2], c[3]};
  v4f hi = {c[4], c[5], c[6], c[7]};
  *(v4f*)(pout)     = lo;   // global_store_b128
  *(v4f*)(pout + 4) = hi;   // global_store_b128
}

// ---------------------------------------------------------------------------
// Kernel 2: ei[bh,n] = Wh[bh,n,:] . a1[h], ej[bh,n] = Wh[bh,n,:] . a2[h]
// Reads WhT: fixed d, consecutive n across threads -> fully coalesced.
// ---------------------------------------------------------------------------
__global__ __launch_bounds__(256) void k_eiej(const float* __restrict__ WhT,
                                              const float* __restrict__ a,
                                              float* __restrict__ ei,
                                              float* __restrict__ ej) {
  const int idx = blockIdx.x * 256 + threadIdx.x;   // bh*N + n
  const int bh = idx >> 10;
  const int n  = idx & (Nc - 1);
  const int h  = bh & (Hc - 1);
  const float* base = WhT + (size_t)bh * HSc * Nc + n;
  const float* a1 = a + h * 2 * HSc;
  const float* a2 = a1 + HSc;
  float s1 = 0.f, s2 = 0.f;
#pragma unroll 8
  for (int d = 0; d < HSc; ++d) {
    float w = base[(size_t)d * Nc];
    s1 += w * a1[d];
    s2 += w * a2[d];
  }
  ei[idx] = s1;
  ej[idx] = s2;
}

// ---------------------------------------------------------------------------
// Kernel 3: per (b,h, 32-row chunk):
//   e_ij = leakyrelu(ei_i + ej_j); masked -> MASK_VAL
//   two-pass softmax (rowmax; unnormalized P=exp(e-m) into LDS; rowsum)
//   out tile = (P @ Wh) * (1/l) via fp32 WMMA, then ELU.
// Block: 256 threads = 8 waves; waves cover 2 i-tiles x 4 d-tiles.
// ej row staged via GLOBAL_LOAD_ASYNC_TO_LDS_B128 (ASYNCcnt path).
// ---------------------------------------------------------------------------
__global__ __launch_bounds__(256) void k_attn(const float* __restrict__ WhT,
                                              const int* __restrict__ adj,
                                              const float* __restrict__ ei,
                                              const float* __restrict__ ej,
                                              float* __restrict__ out) {
  extern __shared__ float smem[];
  float* ej_s = smem;             // 1024 (16B aligned, offset 0)
  float* ei_s = ej_s + Nc;        // 32
  float* m_s  = ei_s + 32;        // 32
  float* l_s  = m_s + 32;         // 32
  float* red  = l_s + 32;         // 256
  float* P    = red + 256;        // 32 * 1024

  const int t = threadIdx.x;
  const int chunks = Nc / 32;                 // 32
  const int bh = blockIdx.x / chunks;
  const int i0 = (blockIdx.x % chunks) * 32;
  const int b = bh >> 2, h = bh & 3;

  // ---- stage ej[bh][0..1023] into LDS: 256 threads x 16B async copies ----
  {
    const float* ejb = ej + (size_t)bh * Nc;
    unsigned loff = (unsigned)(size_t)ej_s + (unsigned)t * 16u;   // LDS byte offset
    unsigned long long gaddr = (unsigned long long)(const void*)(ejb + t * 4);
    asm volatile("global_load_async_to_lds_b128 %0, %1, off"
                 :: "v"(loff), "v"(gaddr) : "memory");
  }
  if (t < 32) ei_s[t] = ei[(size_t)bh * Nc + i0 + t];
  asm volatile("s_wait_asynccnt 0x0" ::: "memory");
  __syncthreads();

  const int i  = t >> 3;        // local row 0..31
  const int s  = t & 7;         // j-slice
  const int j0 = s * 128;
  const float eival = ei_s[i];
  const int* arow = adj + (size_t)(i0 + i) * Nc + j0;

  // ---- pass A: row max -------------------------------------------------
  float mx = MASK_VAL;
  for (int jj = 0; jj < 128; jj += 4) {
    int4 av = *(const int4*)(arow + jj);
    float e0 = eival + ej_s[j0 + jj + 0]; e0 = e0 > 0.f ? e0 : ALPHA * e0; e0 = av.x > 0 ? e0 : MASK_VAL;
    float e1 = eival + ej_s[j0 + jj + 1]; e1 = e1 > 0.f ? e1 : ALPHA * e1; e1 = av.y > 0 ? e1 : MASK_VAL;
    float e2 = eival + ej_s[j0 + jj + 2]; e2 = e2 > 0.f ? e2 : ALPHA * e2; e2 = av.z > 0 ? e2 : MASK_VAL;
    float e3 = eival + ej_s[j0 + jj + 3]; e3 = e3 > 0.f ? e3 : ALPHA * e3; e3 = av.w > 0 ? e3 : MASK_VAL;
    mx = fmaxf(mx, fmaxf(fmaxf(e0, e1), fmaxf(e2, e3)));
  }
  red[t] = mx;
  __syncthreads();
  if (t < 32) {
    float m = red[t * 8];
#pragma unroll
    for (int k = 1; k < 8; ++k) m = fmaxf(m, red[t * 8 + k]);
    m_s[t] = m;
  }
  __syncthreads();

  // ---- pass B: unnormalized P + row sum --------------------------------
  const float m = m_s[i];
  float sum = 0.f;
  float* Prow = P + (size_t)i * Nc + j0;
  for (int jj = 0; jj < 128; jj += 4) {
    int4 av = *(const int4*)(arow + jj);
    float e0 = eival + ej_s[j0 + jj + 0]; e0 = e0 > 0.f ? e0 : ALPHA * e0; e0 = av.x > 0 ? e0 : MASK_VAL;
    float e1 = eival + ej_s[j0 + jj + 1]; e1 = e1 > 0.f ? e1 : ALPHA * e1; e1 = av.y > 0 ? e1 : MASK_VAL;
    float e2 = eival + ej_s[j0 + jj + 2]; e2 = e2 > 0.f ? e2 : ALPHA * e2; e2 = av.z > 0 ? e2 : MASK_VAL;
    float e3 = eival + ej_s[j0 + jj + 3]; e3 = e3 > 0.f ? e3 : ALPHA * e3; e3 = av.w > 0 ? e3 : MASK_VAL;
    float p0 = __expf(e0 - m), p1 = __expf(e1 - m);
    float p2 = __expf(e2 - m), p3 = __expf(e3 - m);
    Prow[jj + 0] = p0; Prow[jj + 1] = p1;
    Prow[jj + 2] = p2; Prow[jj + 3] = p3;
    sum += (p0 + p1) + (p2 + p3);
  }
  red[t] = sum;
  __syncthreads();
  if (t < 32) {
    float l = 0.f;
#pragma unroll
    for (int k = 0; k < 8; ++k) l += red[t * 8 + k];
    l_s[t] = l;
  }
  __syncthreads();

  // ---- phase 2: out = (P @ Wh) / l, ELU -------------------------------
  const int wave  = t >> 5, lane = t & 31;
  const int itile = wave >> 2;        // 0..1
  const int dtile = wave & 3;         // 0..3
  const int lm    = lane & 15;
  const int koff  = (lane < 16) ? 0 : 2;

  // B from WhT: fixed d-column (lane), consecutive K pair -> one b64 load
  const float* pb = WhT + (size_t)bh * HSc * Nc + (size_t)(dtile * 16 + lm) * Nc + koff;
  const float* pA = P + (size_t)(itile * 16 + lm) * Nc + koff;   // A: LDS b64 loads

  v8f c = {};
  for (int jc = 0; jc < Nc; jc += 64) {
    __builtin_prefetch(pb + (size_t)(jc + 64), 0, 1);            // global_prefetch_b8
#pragma unroll
    for (int j = jc; j < jc + 64; j += 4) {
      v2f av = *(const v2f*)(pA + j);                            // ds_load_b64
      v2f bm = *(const v2f*)(pb + j);                            // global_load_b64
      c = __builtin_amdgcn_wmma_f32_16x16x4_f32(false, av, false, bm,
                                                (short)0, c, false, false);
    }
  }

  const int rbase = (lane < 16) ? 0 : 8;
#pragma unroll
  for (int r = 0; r < 8; ++r) {
    const int row = itile * 16 + rbase + r;            // local row in chunk
    const float inv = 1.0f / l_s[row];
    float v = c[r] * inv;
    v = v > 0.f ? v : (__expf(v) - 1.0f);              // ELU
    out[(size_t)(b * Nc + i0 + row) * OUTc + h * HSc + dtile * 16 + lm] = v;
  }
}

// ---------------------------------------------------------------------------
extern "C" void kernel_launch(void* const* d_in, const int* in_sizes, int n_in,
                              void* d_out, int out_size, void* d_ws, size_t ws_size,
                              hipStream_t stream) {
  const float* x   = (const float*)d_in[0];   // (8,1024,256)
  const int*   adj = (const int*)d_in[1];     // (1024,1024)
  const float* W   = (const float*)d_in[2];   // (4,256,64)
  const float* a   = (const float*)d_in[3];   // (4,128)
  float* out = (float*)d_out;                 // (8,1024,256)

  float* WhT = (float*)d_ws;                                  // 8*4*64*1024 floats = 8 MB
  float* ei  = WhT + (size_t)Bc * Hc * Nc * HSc;              // 32768 floats
  float* ej  = ei + (size_t)Bc * Hc * Nc;                     // 32768 floats

  // K1: 8192 tiles, 8 waves/block
  k_wh<<<1024, 256, 0, stream>>>(x, W, WhT);
  // K2: 32768 rows
  k_eiej<<<128, 256, 0, stream>>>(WhT, a, ei, ej);
  // K3: B*H*(N/32) = 1024 blocks; dynamic LDS = (1024+32+32+32+256+32*1024)*4
  const size_t lds = (size_t)(Nc + 32 + 32 + 32 + 256 + 32 * Nc) * sizeof(float);
  k_attn<<<1024, 256, lds, stream>>>(WhT, adj, ei, ej, out);
}